// EdgeConvolution_17506286698576
// MI455X (gfx1250) — compile-verified
//
#include <hip/hip_runtime.h>
#include <float.h>

// Problem constants (reference: B=2, N=512, C=128, UNITS=128)
#define BB     2
#define NN     512
#define CC     128
#define UU     128
#define M_TOT  (BB * NN)          // 1024 rows (b,i)
#define OUTW   (UU + 2 * CC)      // 384 output features per row
#define RA_STRIDE 132             // 128 + 4 pad: avoids LDS bank conflicts
                                  // (row stride !≡ 0 mod 64 banks)

typedef __attribute__((ext_vector_type(2)))  float v2f;
typedef __attribute__((ext_vector_type(8)))  float v8f;

// ---------------------------------------------------------------------------
// Single fused kernel. One 256-thread block (8 wave32) owns 16 rows (b,i):
//
//   Phase A: count1[r] = sum_j adj[m,j]  (wave-per-2-rows, float4/b128 loads,
//            shuffle butterfly), asel[r] = adj[m, xidx[m]] gather,
//            anyB = any_o(bias[o] > 0).
//   Phase B: dual f32 WMMA GEMM (u = X@W1, t = X@W2) — 8 waves = 8 O-tiles of
//            the shared 16-row M-tile; fused epilogue
//            ra = relu(u + (asel-1)*t + bias) -> LDS; per-row any(ra>0).
//   Phase C: n = count1*anyA + (N-count1)*anyB,
//            maxp[o]  = max(count1>0 ? ra : -inf, count1<N ? relu(bias) : -inf)
//            avgpool  = count1 * [x, (asel-1)*x] / n.
//
// V_WMMA_F32_16X16X4_F32 VGPR layouts (ISA 7.12.2):
//   A 16x4: lanes 0-15 -> M=lane, v.x=K0 v.y=K1 ; lanes 16-31 -> v.x=K2 v.y=K3
//   B 4x16: mirrored (N=lane&15; K pair selected by lane>=16)
//   C/D:    VGPR r -> M = r + (lane<16 ? 0 : 8), N = lane&15
// ---------------------------------------------------------------------------
__global__ void __launch_bounds__(256) edgeconv_fused_kernel(
    const float* __restrict__ X,     // [M_TOT, CC]   inputs
    const float* __restrict__ adj,   // [M_TOT, NN]   adjacency ({0,1} floats)
    const int*   __restrict__ xidx,  // [M_TOT]
    const float* __restrict__ W,     // [2*CC, UU]    w1 rows 0..127, w2 rows 128..255
    const float* __restrict__ bias,  // [UU]
    float* __restrict__ out)         // [M_TOT, OUTW]
{
    __shared__ float s_ra[16 * RA_STRIDE]; // relu'd GEMM tile, 16 x 128 (padded)
    __shared__ float s_cnt[16];            // count1 per row
    __shared__ float s_asel[16];           // adj[m, xidx[m]] per row
    __shared__ float s_n[16];              // nonzero-count denominator per row
    __shared__ int   s_anyA[16];           // any_o(ra > 0) per row
    __shared__ int   s_anyB;               // any_o(relu(bias) > 0)

    const int tid  = threadIdx.x;
    const int lane = tid & 31;
    const int wid  = tid >> 5;             // wave 0..7
    const int m0   = blockIdx.x * 16;      // first global row of this M-tile

    // ---- init flags -------------------------------------------------------
    if (tid < 16) s_anyA[tid] = 0;
    if (tid == 16) s_anyB = 0;
    __syncthreads();

    // ---- Phase A: adjacency row sums, asel gather, bias flag --------------
    // Each wave reduces 2 rows; 512 floats/row as 4 x float4 per lane.
    #pragma unroll
    for (int rr = 2 * wid; rr < 2 * wid + 2; ++rr) {
        const float4* arow4 = (const float4*)(adj + (size_t)(m0 + rr) * NN);
        float s = 0.0f;
        #pragma unroll
        for (int q = 0; q < 4; ++q) {
            const float4 v = arow4[lane + 32 * q];  // global_load_b128
            s += v.x + v.y + v.z + v.w;
        }
        #pragma unroll
        for (int off = 16; off >= 1; off >>= 1)
            s += __shfl_xor(s, off, 32);
        if (lane == 0) s_cnt[rr] = s;
    }
    if (tid < 16)
        s_asel[tid] = adj[(size_t)(m0 + tid) * NN + xidx[m0 + tid]];
    if (tid < UU && bias[tid] > 0.0f)
        atomicOr(&s_anyB, 1);
    __syncthreads();

    // ---- Phase B: dual WMMA GEMM + fused epilogue -------------------------
    const int ml    = lane & 15;
    const int khalf = (lane >> 4) << 1;    // 0 or 2  (A/B K-pair select)
    const int mofs  = (lane >> 4) << 3;    // 0 or 8  (C/D row-half select)
    const int o0    = wid * 16;            // this wave's 16-wide O-tile

    const float* __restrict__ w1   = W;
    const float* __restrict__ w2   = W + CC * UU;
    const float* __restrict__ arow = X + (size_t)(m0 + ml) * CC;

    v8f acc1 = {};   // X @ W1 tile
    v8f acc2 = {};   // X @ W2 tile

    #pragma unroll 8
    for (int k = 0; k < CC; k += 4) {
        const v2f a = *(const v2f*)(arow + k + khalf);     // global_load_b64
        v2f b1, b2;
        b1.x = w1[(size_t)(k + khalf    ) * UU + o0 + ml];
        b1.y = w1[(size_t)(k + khalf + 1) * UU + o0 + ml];
        b2.x = w2[(size_t)(k + khalf    ) * UU + o0 + ml];
        b2.y = w2[(size_t)(k + khalf + 1) * UU + o0 + ml];
        acc1 = __builtin_amdgcn_wmma_f32_16x16x4_f32(
                   false, a, false, b1, (short)0, acc1, false, false);
        acc2 = __builtin_amdgcn_wmma_f32_16x16x4_f32(
                   false, a, false, b2, (short)0, acc2, false, false);
    }

    // Epilogue: ra = relu(u + (asel-1)*t + bias) -> LDS, per-row any-flag.
    const float bo = bias[o0 + ml];
    #pragma unroll
    for (int r = 0; r < 8; ++r) {
        const int   mlr = r + mofs;        // local row 0..15
        const float s   = acc1[r] + (s_asel[mlr] - 1.0f) * acc2[r] + bo;
        const float ra  = fmaxf(s, 0.0f);
        s_ra[mlr * RA_STRIDE + o0 + ml] = ra;
        if (ra > 0.0f) atomicOr(&s_anyA[mlr], 1);
    }
    __syncthreads();

    // ---- Phase C: denominators, then pooled outputs -----------------------
    if (tid < 16) {
        const float c1 = s_cnt[tid];
        const float c0 = (float)NN - c1;
        s_n[tid] = c1 * (s_anyA[tid] ? 1.0f : 0.0f)
                 + c0 * (s_anyB      ? 1.0f : 0.0f);
    }
    __syncthreads();

    // maxp: 16 rows x 128 outputs
    for (int idx = tid; idx < 16 * UU; idx += 256) {
        const int   row = idx >> 7;
        const int   o   = idx & (UU - 1);
        const float c1  = s_cnt[row];
        const float ra  = s_ra[row * RA_STRIDE + o];
        const float r0  = fmaxf(bias[o], 0.0f);
        const float ma  = (c1 > 0.0f)      ? ra : -FLT_MAX;  // edges with a=1
        const float mz  = (c1 < (float)NN) ? r0 : -FLT_MAX;  // edges with a=0
        out[(size_t)(m0 + row) * OUTW + o] = fmaxf(ma, mz);
    }

    // avgpool: sum_j edging = count1 * [x, (asel-1)*x];  divide by n
    for (int idx = tid; idx < 16 * CC; idx += 256) {
        const int   row = idx >> 7;
        const int   c   = idx & (CC - 1);
        const float x   = X[(size_t)(m0 + row) * CC + c];
        const float c1  = s_cnt[row];
        const float inv = 1.0f / s_n[row];
        float* orow = out + (size_t)(m0 + row) * OUTW;
        orow[UU + c]      = c1 * x * inv;
        orow[UU + CC + c] = c1 * (s_asel[row] - 1.0f) * x * inv;
    }
}

// ---------------------------------------------------------------------------
// Launch: one kernel, 64 blocks x 256 threads (8 wave32 per block).
// ---------------------------------------------------------------------------
extern "C" void kernel_launch(void* const* d_in, const int* in_sizes, int n_in,
                              void* d_out, int out_size, void* d_ws, size_t ws_size,
                              hipStream_t stream) {
    const float* X    = (const float*)d_in[0];  // inputs     [2,512,128]
    const float* adj  = (const float*)d_in[1];  // adj_matrix [2,512,512]
    const int*   xidx = (const int*)  d_in[2];  // xidx       [2,512]
    const float* W    = (const float*)d_in[3];  // w          [1,256,128]
    const float* bias = (const float*)d_in[4];  // b          [128]
    float* out = (float*)d_out;                 // [2,512,384]

    edgeconv_fused_kernel<<<M_TOT / 16, 256, 0, stream>>>(
        X, adj, xidx, W, bias, out);
}